// MultiscalePatchExtractor_39599598469255
// MI455X (gfx1250) — compile-verified
//
#include <hip/hip_runtime.h>
#include <hip/hip_bf16.h>

// ---------------------------------------------------------------------------
// PatchExtractor.forward_indexes on gfx1250 (MI455X), fp32 WMMA path.
//
// Shapes: x (32,512,512,3) f32 | W (48,256) f32 | b (256) f32 |
//         indexes (32,8192) i32 | h_offset (32) i32 | w_offset (32) i32
// Out:    emb (32,8192,256) f32 ++ mask (32,8192) f32  (concatenated flat)
//
// Key simplification: 'available' indexes are a contiguous 128x128 rectangle
// of the 256x256 template grid, so the scatter/gather hash table inverts
// analytically: a_idx = (r/256 - hoff_p)*128 + (r%256 - woff_p) when in-rect.
// ---------------------------------------------------------------------------

typedef __attribute__((ext_vector_type(2))) float v2f;
typedef __attribute__((ext_vector_type(8))) float v8f;

#define BATCH   32
#define BREQ    8192
#define CEMB    256
#define DIN     48        // 4*4*3
#define NIDX    65536     // 256*256 template grid
#define ROWS_PER_BLOCK 128
#define WAVES   8

// LDS layout (floats):
//   ldsW     : 12288  (6144 float2 pre-swizzled B-fragments: [k=0..11][n=0..15][lane=0..31])
//   ldsPatch :  6144  (8 waves * 16 rows * 48 floats, A staging)
//   ldsBias  :   256
//   ldsMask  :   128  (per-row mask as float)
//   ldsBase  :   128  (per-row global base offset of gathered patch, int)
#define LDS_W_F     12288
#define LDS_PATCH_F 6144
#define LDS_FLOATS  (LDS_W_F + LDS_PATCH_F + 256 + 128 + 128)
#define LDS_BYTES   (LDS_FLOATS * 4)   // 75776 B

__global__ __launch_bounds__(256)
void patch_embed_wmma_f32(const float* __restrict__ x,
                          const float* __restrict__ W,
                          const float* __restrict__ bias,
                          const int*   __restrict__ indexes,
                          const int*   __restrict__ h_offset,
                          const int*   __restrict__ w_offset,
                          float*       __restrict__ out)
{
    extern __shared__ float smem[];
    float* ldsW     = smem;                       // swizzled W, float2 granularity
    float* ldsPatch = ldsW + LDS_W_F;             // A staging
    float* ldsBias  = ldsPatch + LDS_PATCH_F;
    float* ldsMask  = ldsBias + 256;
    int*   ldsBase  = (int*)(ldsMask + 128);

    const int tid   = threadIdx.x;
    const int lane  = tid & 31;
    const int wave  = tid >> 5;
    const int batch = blockIdx.x >> 6;                 // 64 blocks per batch
    const int rowBase = (blockIdx.x & 63) * ROWS_PER_BLOCK;

    // ---- Phase 1: stage W into LDS, pre-swizzled into per-lane B-fragments.
    // Entry e = (k*16 + n)*32 + lane holds float2:
    //   lane<16 : ( W[4k+0][n*16+lane],    W[4k+1][n*16+lane]    )
    //   lane>=16: ( W[4k+2][n*16+lane-16], W[4k+3][n*16+lane-16] )
    for (int e = tid; e < LDS_W_F / 2; e += 256) {
        const int el   = e & 31;
        const int n    = (e >> 5) & 15;
        const int k    = e >> 9;
        const int row0 = 4 * k + 2 * (el >> 4);
        const int col  = n * 16 + (el & 15);
        ldsW[2 * e + 0] = W[row0 * CEMB + col];
        ldsW[2 * e + 1] = W[(row0 + 1) * CEMB + col];
    }
    ldsBias[tid] = bias[tid];   // 256 threads, 256 cols

    // ---- Phase 2: per-row metadata (threads 0..127), analytic table inversion.
    if (tid < ROWS_PER_BLOCK) {
        const int j   = rowBase + tid;
        const int idx = indexes[batch * BREQ + j];
        const int hp  = h_offset[batch] >> 2;          // offset in patches
        const int wp  = w_offset[batch] >> 2;
        const bool valid = (idx >= 0) && (idx < NIDX);
        const int r  = idx >> 8;
        const int c  = idx & 255;
        const int rh = r - hp;
        const int rw = c - wp;
        const bool inr = valid && (rh >= 0) && (rh < 128) && (rw >= 0) && (rw < 128);
        const int a_idx = inr ? (rh * 128 + rw) : j;   // fallback = j % A, j < A
        const int ph = a_idx >> 7;
        const int pw = a_idx & 127;
        // base offset (floats) of pixel (ph*4, pw*4) in NHWC image
        ldsBase[tid] = ((batch * 512 + ph * 4) * 512 + pw * 4) * 3;
        const float mf = inr ? 1.0f : 0.0f;
        ldsMask[tid] = mf;
        out[(size_t)BATCH * BREQ * CEMB + (size_t)batch * BREQ + j] = mf;
    }
    __syncthreads();

    // ---- Phase 3: gather this wave's 16 patches (16 x 48 f32) into LDS.
    // A patch is 4 image rows of 12 contiguous floats (NHWC, stride 512*3).
    {
        const int pbase = wave * 16;
        for (int t = lane; t < 16 * DIN; t += 32) {
            const int p   = t / DIN;
            const int d   = t - p * DIN;
            const int i   = d / 12;          // patch row
            const int rem = d - i * 12;      // contiguous within image row
            ldsPatch[(pbase + p) * DIN + d] = x[ldsBase[pbase + p] + i * 1536 + rem];
        }
    }
    __syncthreads();

    // ---- Phase 4: 12-step chained V_WMMA_F32_16X16X4_F32 per 16x16 col tile.
    const int half = lane >> 4;          // K-pair select within fragment
    const int lm   = lane & 15;

    // A-fragments: lane holds (patchRow=lm, K=4k+2*half .. +1), 8B-aligned pairs.
    v2f afrag[12];
    {
        const float* pA = ldsPatch + (wave * 16 + lm) * DIN + 2 * half;
#pragma unroll
        for (int k = 0; k < 12; ++k)
            afrag[k] = *(const v2f*)(pA + 4 * k);
    }
    // Per-row mask for this lane's D rows (M = v + 8*half).
    float mrow[8];
#pragma unroll
    for (int v = 0; v < 8; ++v)
        mrow[v] = ldsMask[wave * 16 + v + 8 * half];

    const int j16 = rowBase + wave * 16;
    float* outBase = out + ((size_t)(batch * BREQ + j16)) * CEMB;

#pragma unroll
    for (int n = 0; n < 16; ++n) {
        v8f acc = {};
        const float* pB = ldsW + (n * 32 + lane) * 2;
#pragma unroll
        for (int k = 0; k < 12; ++k) {
            const v2f bfrag = *(const v2f*)(pB + k * (16 * 32 * 2));
            acc = __builtin_amdgcn_wmma_f32_16x16x4_f32(
                /*neg_a=*/false, afrag[k],
                /*neg_b=*/false, bfrag,
                /*c_mod=*/(short)0, acc,
                /*reuse_a=*/false, /*reuse_b=*/false);
        }
        const int col = n * 16 + lm;
        const float bv = ldsBias[col];
        float* po = outBase + col;
#pragma unroll
        for (int v = 0; v < 8; ++v) {
            const int M = v + 8 * half;
            po[(size_t)M * CEMB] = (acc[v] + bv) * mrow[v];
        }
    }
}

extern "C" void kernel_launch(void* const* d_in, const int* in_sizes, int n_in,
                              void* d_out, int out_size, void* d_ws, size_t ws_size,
                              hipStream_t stream) {
    (void)in_sizes; (void)n_in; (void)d_ws; (void)ws_size; (void)out_size;
    const float* x    = (const float*)d_in[0];
    const float* W    = (const float*)d_in[1];
    const float* bias = (const float*)d_in[2];
    const int*   idx  = (const int*)d_in[3];
    const int*   hoff = (const int*)d_in[4];
    const int*   woff = (const int*)d_in[5];
    float* out = (float*)d_out;

    const int blocks = BATCH * (BREQ / ROWS_PER_BLOCK);   // 32 * 64 = 2048
    patch_embed_wmma_f32<<<blocks, 256, LDS_BYTES, stream>>>(
        x, W, bias, idx, hoff, woff, out);
}